// GCN_57939108823477
// MI455X (gfx1250) — compile-verified
//
#include <hip/hip_runtime.h>

typedef float v2f __attribute__((ext_vector_type(2)));
typedef float v8f __attribute__((ext_vector_type(8)));
typedef unsigned int u32x4 __attribute__((ext_vector_type(4)));
typedef int i32x4 __attribute__((ext_vector_type(4)));
typedef int i32x8 __attribute__((ext_vector_type(8)));

#define NEG_SLOPE 0.1f
#define LRELU(v) ((v) > 0.0f ? (v) : NEG_SLOPE * (v))

static inline int cdiv(long long a, long long b) { return (int)((a + b - 1) / b); }

// ---------------------------------------------------------------- zero fill
__global__ void k_zero(float* __restrict__ p, size_t n) {
  size_t i4 = ((size_t)blockIdx.x * blockDim.x + threadIdx.x) * 4;
  if (i4 + 3 < n) {
    *(float4*)(p + i4) = make_float4(0.f, 0.f, 0.f, 0.f);
  } else {
    for (size_t j = i4; j < n; ++j) p[j] = 0.f;
  }
}

// ------------------------------------------------- embed layer 1 (K=3 or 6)
__global__ void k_embed1(const float* __restrict__ x, const float* __restrict__ W1,
                         const float* __restrict__ b1, float* __restrict__ h1,
                         int n, int K) {
  int idx = blockIdx.x * blockDim.x + threadIdx.x;
  if (idx >= n * 128) return;
  int node = idx >> 7, c = idx & 127;
  float acc = b1[c];
  const float* xr = x + (size_t)node * K;
  for (int k = 0; k < K; ++k) acc = fmaf(xr[k], W1[k * 128 + c], acc);
  h1[idx] = LRELU(acc);
}

// ---------------------------------------------- fp32 WMMA GEMM [n,128]x[128,128]
// One wave = 16 output rows x 128 output cols; 4 waves per block.
// A-tile staged into LDS by the Tensor Data Mover (tensor_load_to_lds) with
// hardware LDS padding 128+4 dwords/row (bank-spread stride 132).
// Epilogue options: bias[col], extra[row,col]*invdeg[row] (SAGE agg), LeakyReLU.
template <int HAS_BIAS, int HAS_EXTRA, int RELU>
__global__ __launch_bounds__(128) void k_gemm128(
    const float* __restrict__ A, const float* __restrict__ W,
    const float* __restrict__ bias, const float* __restrict__ extra,
    const float* __restrict__ invdeg, float* __restrict__ out, int n) {
  __shared__ float lds[4 * 16 * 132];
  const int wv = threadIdx.x >> 5;
  const int lane = threadIdx.x & 31;
  const int waveRow = blockIdx.x * 64 + wv * 16;
  float* la = lds + wv * (16 * 132);

  // ---- TDM: DMA this wave's 16x128 fp32 tile global->LDS (wave-uniform D#)
  int gRow = waveRow;
  if (gRow >= n) gRow = (n > 16) ? (n - 16) : 0;  // fully-OOB wave: load valid rows, never stored
  int rows = n - gRow;
  if (rows > 16) rows = 16;
  unsigned long long gaddr = (unsigned long long)(const void*)(A + (size_t)gRow * 128);

  u32x4 g0;
  g0.x = 1u;                                  // count=1, user descriptor
  g0.y = (unsigned)(size_t)(void*)la;         // lds_addr (bytes, wave-relative)
  g0.z = (unsigned)(gaddr & 0xffffffffull);   // global_addr[31:0]
  g0.w = (unsigned)((gaddr >> 32) & 0x01ffffffull) | (2u << 30);  // addr[56:32] | type=2

  i32x8 g1;
  g1[0] = (int)((2u << 16)      // data_size = 4 bytes
              | (1u << 20)      // pad_enable
              | (6u << 22)      // pad_interval: 128 dwords
              | (3u << 25));    // pad_amount: 4 dwords -> row stride 132
  g1[1] = (int)(128u << 16);                        // tensor_dim0 = 128
  g1[2] = (int)(((unsigned)rows & 0xffffu) << 16);  // tensor_dim1 = rows
  g1[3] = (int)(128u << 16);                        // tile_dim0 = 128
  g1[4] = rows;                                     // tile_dim1 = rows
  g1[5] = 128;                                      // tensor_dim0_stride = 128
  g1[6] = 0;
  g1[7] = 0;
  i32x4 gz4 = {0, 0, 0, 0};                         // groups 2/3: tile_dim2..4 unused
  i32x8 gz8 = {0, 0, 0, 0, 0, 0, 0, 0};             // trailing group (clang-23 6-arg form)

  __builtin_amdgcn_tensor_load_to_lds(g0, g1, gz4, gz4, gz8, 0);
  __builtin_amdgcn_s_wait_tensorcnt((short)0);
  __syncthreads();
  asm volatile("" ::: "memory");  // TDM wrote LDS behind the compiler's back

  const int m = lane & 15;          // A: row / B: col / D: col
  const int kk = (lane >> 4) << 1;  // K sub-offset (0 or 2) per half-wave
  v8f acc[8] = {};

  for (int kb = 0; kb < 128; kb += 4) {
    v2f a;
    a.x = la[m * 132 + kb + kk];
    a.y = la[m * 132 + kb + kk + 1];
    const float* w0 = W + (size_t)(kb + kk) * 128;
#pragma unroll
    for (int t = 0; t < 8; ++t) {
      v2f b;
      b.x = w0[t * 16 + m];
      b.y = w0[128 + t * 16 + m];
      acc[t] = __builtin_amdgcn_wmma_f32_16x16x4_f32(
          false, a, false, b, (short)0, acc[t], false, false);
    }
  }

#pragma unroll
  for (int t = 0; t < 8; ++t) {
    int col = t * 16 + m;
#pragma unroll
    for (int r = 0; r < 8; ++r) {
      int row = waveRow + r + ((lane >> 4) << 3);  // lanes>=16 hold M=r+8
      if (row < n) {
        float v = acc[t][r];
        if (HAS_BIAS) v += bias[col];
        if (HAS_EXTRA) v = fmaf(extra[(size_t)row * 128 + col], invdeg[row], v);
        if (RELU) v = LRELU(v);
        out[(size_t)row * 128 + col] = v;
      }
    }
  }
}

// ------------------------------------------------------- x = xcat[object_ptv]
__global__ void k_gather(const float* __restrict__ xcat, const int* __restrict__ ptv,
                         float* __restrict__ h, int N) {
  int idx = blockIdx.x * blockDim.x + threadIdx.x;
  if (idx >= N * 32) return;
  int node = idx >> 5, c4 = (idx & 31) << 2;
  int s = ptv[node];
  *(float4*)(h + (size_t)node * 128 + c4) =
      *(const float4*)(xcat + (size_t)s * 128 + c4);
}

// ------------------------------------------------------------------- degree
__global__ void k_deg(const int* __restrict__ dst, float* __restrict__ deg, int nE) {
  int e = blockIdx.x * blockDim.x + threadIdx.x;
  if (e < nE) atomicAdd(deg + dst[e], 1.0f);
}
__global__ void k_invdeg(float* __restrict__ deg, int N) {
  int i = blockIdx.x * blockDim.x + threadIdx.x;
  if (i < N) deg[i] = 1.0f / fmaxf(deg[i], 1.0f);
}

// ------------------------------------------- agg[dst] += p[src] (128 floats)
__global__ void k_scatter_agg(const float* __restrict__ p, const int* __restrict__ src,
                              const int* __restrict__ dst, float* __restrict__ agg,
                              int nE) {
  long long idx = (long long)blockIdx.x * blockDim.x + threadIdx.x;
  if (idx >= (long long)nE * 32) return;
  int e = (int)(idx >> 5);
  int c4 = ((int)idx & 31) << 2;
  int s = src[e], d = dst[e];
  float4 v = *(const float4*)(p + (size_t)s * 128 + c4);
  float* a = agg + (size_t)d * 128 + c4;
  atomicAdd(a + 0, v.x);
  atomicAdd(a + 1, v.y);
  atomicAdd(a + 2, v.z);
  atomicAdd(a + 3, v.w);
}

// --------------------------------- output head: yl = h.Wl_o, yr = h.Wr_o (wave/node)
__global__ void k_head(const float* __restrict__ h, const float* __restrict__ Wl,
                       const float* __restrict__ Wr, float* __restrict__ yl,
                       float* __restrict__ yr, int N) {
  int node = (blockIdx.x * blockDim.x + threadIdx.x) >> 5;
  int lane = threadIdx.x & 31;
  if (node >= N) return;
  float4 hv = *(const float4*)(h + (size_t)node * 128 + lane * 4);
  float4 wl = *(const float4*)(Wl + lane * 4);
  float4 wr = *(const float4*)(Wr + lane * 4);
  float sl = hv.x * wl.x + hv.y * wl.y + hv.z * wl.z + hv.w * wl.w;
  float sr = hv.x * wr.x + hv.y * wr.y + hv.z * wr.z + hv.w * wr.w;
  for (int o = 16; o > 0; o >>= 1) {
    sl += __shfl_xor(sl, o, 32);
    sr += __shfl_xor(sr, o, 32);
  }
  if (lane == 0) { yl[node] = sl; yr[node] = sr; }
}

__global__ void k_scatter_s(const float* __restrict__ yl, const int* __restrict__ src,
                            const int* __restrict__ dst, float* __restrict__ aggs,
                            int nE) {
  int e = blockIdx.x * blockDim.x + threadIdx.x;
  if (e < nE) atomicAdd(aggs + dst[e], yl[src[e]]);
}

__global__ void k_final(const float* __restrict__ aggs, const float* __restrict__ invdeg,
                        const float* __restrict__ yr, const float* __restrict__ bl_o,
                        float* __restrict__ out, int N) {
  int i = blockIdx.x * blockDim.x + threadIdx.x;
  if (i >= N) return;
  float z = fmaf(aggs[i], invdeg[i], bl_o[0] + yr[i]);
  out[i] = 1.0f / (1.0f + __expf(-z));
}

// ============================================================== host driver
extern "C" void kernel_launch(void* const* d_in, const int* in_sizes, int n_in,
                              void* d_out, int out_size, void* d_ws, size_t ws_size,
                              hipStream_t stream) {
  const float* x_gen  = (const float*)d_in[0];
  const float* x_load = (const float*)d_in[1];
  const float* x_or   = (const float*)d_in[2];
  const float* x_ex   = (const float*)d_in[3];
  const int*   ei     = (const int*)d_in[4];
  const int*   ptv    = (const int*)d_in[5];

  const int nG = in_sizes[0] / 3, nL = in_sizes[1] / 3;
  const int nO = in_sizes[2] / 6, nX = in_sizes[3] / 6;
  const int nE = in_sizes[4] / 2;
  const int N  = in_sizes[5];
  const int* src = ei;
  const int* dst = ei + nE;

  const float* Wl_h = (const float*)d_in[22];
  const float* bl_h = (const float*)d_in[23];
  const float* Wr_h = (const float*)d_in[24];
  const float* Wl_o = (const float*)d_in[25];
  const float* bl_o = (const float*)d_in[26];
  const float* Wr_o = (const float*)d_in[27];

  // workspace layout (floats)
  float* ws = (float*)d_ws;
  const size_t NB = (size_t)N * 128;
  int embMax = nG; if (nL > embMax) embMax = nL; if (nO > embMax) embMax = nO; if (nX > embMax) embMax = nX;
  float* xcat = ws;            // [N,128]; reused as agg after gather
  float* hA   = xcat + NB;     // [N,128]
  float* hB   = hA + NB;       // [N,128]  (p, then next-layer h)
  float* h1   = hB + NB;       // [embMax,128]
  float* dinv = h1 + (size_t)embMax * 128;  // deg -> inv_deg [N]
  float* yl   = dinv + N;
  float* yr   = yl + N;
  float* aggs = yr + N;

  // ---- per-type embeddings -> xcat
  const float* embX[4]  = {x_gen, x_load, x_or, x_ex};
  const int    embN[4]  = {nG, nL, nO, nX};
  const int    embK[4]  = {3, 3, 6, 6};
  const float* embW1[4] = {(const float*)d_in[6],  (const float*)d_in[10],
                           (const float*)d_in[14], (const float*)d_in[18]};
  const float* embB1[4] = {(const float*)d_in[7],  (const float*)d_in[11],
                           (const float*)d_in[15], (const float*)d_in[19]};
  const float* embW2[4] = {(const float*)d_in[8],  (const float*)d_in[12],
                           (const float*)d_in[16], (const float*)d_in[20]};
  const float* embB2[4] = {(const float*)d_in[9],  (const float*)d_in[13],
                           (const float*)d_in[17], (const float*)d_in[21]};
  size_t off = 0;
  for (int t = 0; t < 4; ++t) {
    k_embed1<<<cdiv((long long)embN[t] * 128, 256), 256, 0, stream>>>(
        embX[t], embW1[t], embB1[t], h1, embN[t], embK[t]);
    k_gemm128<1, 0, 1><<<cdiv(embN[t], 64), 128, 0, stream>>>(
        h1, embW2[t], embB2[t], nullptr, nullptr, xcat + off * 128, embN[t]);
    off += (size_t)embN[t];
  }

  // ---- permutation gather + degree
  k_gather<<<cdiv((long long)N * 32, 256), 256, 0, stream>>>(xcat, ptv, hA, N);
  k_zero<<<cdiv((long long)N, 1024), 256, 0, stream>>>(dinv, (size_t)N);
  k_deg<<<cdiv(nE, 256), 256, 0, stream>>>(dst, dinv, nE);
  k_invdeg<<<cdiv(N, 256), 256, 0, stream>>>(dinv, N);

  // ---- 3 hidden SAGE layers
  for (int l = 0; l < 3; ++l) {
    const float* Wl = Wl_h + (size_t)l * 128 * 128;
    const float* bl = bl_h + (size_t)l * 128;
    const float* Wr = Wr_h + (size_t)l * 128 * 128;
    // p = h @ Wl
    k_gemm128<0, 0, 0><<<cdiv(N, 64), 128, 0, stream>>>(
        hA, Wl, nullptr, nullptr, nullptr, hB, N);
    // agg[dst] += p[src]   (agg buffer is L2-resident: 51 MB < 192 MB)
    k_zero<<<cdiv((long long)NB, 1024), 256, 0, stream>>>(xcat, NB);
    k_scatter_agg<<<cdiv((long long)nE * 32, 256), 256, 0, stream>>>(
        hB, src, dst, xcat, nE);
    // h' = lrelu(agg*invdeg + bl + h @ Wr)   (written into hB, then swap)
    k_gemm128<1, 1, 1><<<cdiv(N, 64), 128, 0, stream>>>(
        hA, Wr, bl, xcat, dinv, hB, N);
    float* tmp = hA; hA = hB; hB = tmp;
  }

  // ---- output head (project-then-aggregate: scalar edge traffic)
  k_head<<<cdiv((long long)N * 32, 256), 256, 0, stream>>>(hA, Wl_o, Wr_o, yl, yr, N);
  k_zero<<<cdiv((long long)N, 1024), 256, 0, stream>>>(aggs, (size_t)N);
  k_scatter_s<<<cdiv(nE, 256), 256, 0, stream>>>(yl, src, dst, aggs, nE);
  k_final<<<cdiv(N, 256), 256, 0, stream>>>(aggs, dinv, yr, bl_o, (float*)d_out, N);
}